// BertCRFSoftLexicon_47373489275349
// MI455X (gfx1250) — compile-verified
//
#include <hip/hip_runtime.h>
#include <hip/hip_bf16.h>
#include <math.h>

#define Bn   64
#define Sn   512
#define Hn   768
#define WEn  4
#define MLn  5
#define Dn   50
#define Tn   21
#define KTOT (Hn + WEn * Dn)   /* 968 */
#define KPAD 972               /* 968 + 4: bank-skew (12 mod 64) + 16B align */
#define NTOK (Bn * Sn)         /* 32768 */
#define LEXW (WEn * Dn)        /* 200 */

typedef __attribute__((ext_vector_type(2))) float v2f;
typedef __attribute__((ext_vector_type(8))) float v8f;

// ---------------------------------------------------------------------------
// Fused: stage full 16x968 feats tile in LDS (seq copy + lex gather),
// then [16x968] x [968x32] fp32 WMMA GEMM (two 16-wide N tiles, one per wave).
// Block = 64 threads = 2 waves; grid.x = NTOK/16 = 2048 M-tiles.
// ---------------------------------------------------------------------------
__global__ __launch_bounds__(64) void fused_lex_gemm_kernel(
    const float* __restrict__ seq,       // [NTOK, Hn]
    const int*   __restrict__ lex_ids,   // [NTOK, WEn*MLn]
    const float* __restrict__ lex_wt,    // [NTOK, WEn*MLn]
    const float* __restrict__ word_emb,  // [V, Dn]
    const float* __restrict__ Wm,        // [KTOT, Tn]
    const float* __restrict__ bias,      // [Tn]
    float* __restrict__ logits)          // [NTOK, Tn]
{
    __shared__ float featTile[16 * KPAD];      // 62.2 KB
    __shared__ int   ids_s[16 * WEn * MLn];    // 320 ints
    __shared__ float wt_s [16 * WEn * MLn];    // 320 floats

    const int tid  = threadIdx.x;              // 0..63
    const int row0 = blockIdx.x * 16;

    // Stage per-tile lexicon ids/weights into LDS.
    for (int i = tid; i < 16 * WEn * MLn; i += 64) {
        int r = i / (WEn * MLn);
        int c = i % (WEn * MLn);
        size_t g = (size_t)(row0 + r) * (WEn * MLn) + c;
        ids_s[i] = lex_ids[g];
        wt_s [i] = lex_wt [g];
    }

    // Stage seq part of the A tile: 16 rows x 768 floats, float4 vectorized.
    for (int c = tid; c < 16 * (Hn / 4); c += 64) {
        int r   = c / (Hn / 4);
        int off = (c % (Hn / 4)) * 4;
        const float4 v = *(const float4*)(seq + (size_t)(row0 + r) * Hn + off);
        *(float4*)(&featTile[r * KPAD + off]) = v;
    }
    __syncthreads();

    // Lex gather + weighted sum into the tail 200 columns of each A row.
    for (int i = tid; i < 16 * LEXW; i += 64) {
        int r = i / LEXW;
        int o = i % LEXW;
        int w = o / Dn;
        int d = o % Dn;
        float acc = 0.f;
#pragma unroll
        for (int l = 0; l < MLn; ++l) {
            int   id = ids_s[r * (WEn * MLn) + w * MLn + l];
            float wt = wt_s [r * (WEn * MLn) + w * MLn + l];
            acc += word_emb[(size_t)id * Dn + d] * wt;
        }
        featTile[r * KPAD + Hn + o] = acc;
    }
    __syncthreads();

    const int wave  = tid >> 5;                // N-tile select
    const int lane  = tid & 31;
    const int mrow  = lane & 15;               // A: local M row
    const int khalf = (lane >> 4) << 1;        // lanes 16-31 carry K+2,K+3
    const int col   = wave * 16 + (lane & 15); // global N
    // Columns >= Tn are never stored; B-lane j only affects C column j, so
    // clamping (not zero-padding) is sufficient and branch-free.
    const int colc  = (col < Tn) ? col : (Tn - 1);

    const float* __restrict__ Arow = &featTile[mrow * KPAD + khalf];
    const float* __restrict__ Wc   = Wm + colc;

    v8f acc = {};
#pragma unroll 4
    for (int k = 0; k < KTOT; k += 4) {
        v2f a, bm;
        a.x  = Arow[k];                        // ds_load_b64
        a.y  = Arow[k + 1];
        bm.x = Wc[(size_t)(k + khalf)     * Tn];
        bm.y = Wc[(size_t)(k + khalf + 1) * Tn];
        acc = __builtin_amdgcn_wmma_f32_16x16x4_f32(
            false, a, false, bm, (short)0, acc, false, false);
    }

    // Epilogue: C tile layout — VGPR v: lanes 0-15 => M=v, lanes 16-31 => M=v+8
    if (col < Tn) {
        const float bv   = bias[col];
        const int   rofs = (lane >> 4) << 3;
#pragma unroll
        for (int v = 0; v < 8; ++v) {
            size_t row = (size_t)(row0 + v + rofs);
            logits[row * Tn + col] = acc[v] + bv;
        }
    }
}

// ---------------------------------------------------------------------------
// CRF NLL: one wave32 block per batch. Lane j (< Tn) owns alpha_j.
// Writes per-batch (logZ - num) into loss_part[b].
// ---------------------------------------------------------------------------
__global__ __launch_bounds__(32) void crf_kernel(
    const float* __restrict__ logits,    // [Bn, Sn, Tn] (read from d_out)
    const int*   __restrict__ tags,      // [Bn, Sn]
    const int*   __restrict__ mask,      // [Bn, Sn]
    const float* __restrict__ start_t,   // [Tn]
    const float* __restrict__ end_t,     // [Tn]
    const float* __restrict__ trans,     // [Tn, Tn]
    float* __restrict__ loss_part)       // [Bn]
{
    __shared__ float trans_s[Tn * Tn];
    __shared__ float alpha_s[32];

    const int b    = blockIdx.x;
    const int lane = threadIdx.x;
    const float NEG = -3.0e38f;

    for (int i = lane; i < Tn * Tn; i += 32) trans_s[i] = trans[i];
    __syncthreads();

    const float* em = logits + (size_t)b * Sn * Tn;
    const int*   tg = tags   + (size_t)b * Sn;
    const int*   mk = mask   + (size_t)b * Sn;

    // ---- numerator (parallel over t) + mask sum ----
    float num  = 0.f;
    int   msum = 0;
    for (int t = lane; t < Sn; t += 32) {
        msum += mk[t];
        if (t >= 1) {
            float m = (float)mk[t];
            num += (trans_s[tg[t - 1] * Tn + tg[t]] + em[t * Tn + tg[t]]) * m;
        }
    }
    for (int off = 16; off > 0; off >>= 1) {
        num  += __shfl_down(num, off, 32);
        msum += __shfl_down(msum, off, 32);
    }

    // ---- forward scan (logZ) ----
    float alpha = (lane < Tn) ? (start_t[lane] + em[lane]) : NEG;
    for (int t = 1; t < Sn; ++t) {
        alpha_s[lane] = alpha;
        __syncthreads();
        const float m = (float)mk[t];
        float mx = NEG;
#pragma unroll 1
        for (int i = 0; i < Tn; ++i) {
            float v = alpha_s[i] + ((lane < Tn) ? trans_s[i * Tn + lane] : 0.f);
            mx = fmaxf(mx, v);
        }
        float s = 0.f;
#pragma unroll 1
        for (int i = 0; i < Tn; ++i) {
            float v = alpha_s[i] + ((lane < Tn) ? trans_s[i * Tn + lane] : 0.f);
            s += __expf(v - mx);
        }
        float nxt = mx + __logf(s) + ((lane < Tn) ? em[t * Tn + lane] : 0.f);
        alpha = (m > 0.f && lane < Tn) ? nxt : alpha;
        __syncthreads();
    }

    // logZ = logsumexp_j(alpha_j + end_j)
    float v  = (lane < Tn) ? (alpha + end_t[lane]) : NEG;
    float mx = v;
    for (int off = 16; off > 0; off >>= 1) mx = fmaxf(mx, __shfl_down(mx, off, 32));
    mx = __shfl(mx, 0, 32);
    float s = (lane < Tn) ? __expf(v - mx) : 0.f;
    for (int off = 16; off > 0; off >>= 1) s += __shfl_down(s, off, 32);

    if (lane == 0) {
        float logZ = mx + __logf(s);
        int   last = msum - 1;
        num += start_t[tg[0]] + em[tg[0]] + end_t[tg[last]];
        loss_part[b] = logZ - num;   // contribution to -sum(num - logZ)
    }
}

__global__ __launch_bounds__(64) void loss_reduce_kernel(
    const float* __restrict__ loss_part, float* __restrict__ out_loss)
{
    __shared__ float sdata[64];
    int t = threadIdx.x;
    sdata[t] = loss_part[t];
    __syncthreads();
    for (int off = 32; off > 0; off >>= 1) {
        if (t < off) sdata[t] += sdata[t + off];
        __syncthreads();
    }
    if (t == 0) *out_loss = sdata[0];
}

extern "C" void kernel_launch(void* const* d_in, const int* in_sizes, int n_in,
                              void* d_out, int out_size, void* d_ws, size_t ws_size,
                              hipStream_t stream) {
    const float* seq      = (const float*)d_in[0];   // sequence_output [B,S,H]
    const int*   lex_ids  = (const int*)  d_in[1];   // [B,S,WE,ML]
    const float* lex_wt   = (const float*)d_in[2];   // [B,S,WE,ML]
    const int*   label    = (const int*)  d_in[3];   // [B,S]
    const int*   amask    = (const int*)  d_in[4];   // [B,S]
    const float* word_emb = (const float*)d_in[5];   // [V,D]
    const float* Wm       = (const float*)d_in[6];   // [968,21]
    const float* bias     = (const float*)d_in[7];   // [21]
    const float* start_t  = (const float*)d_in[8];   // [21]
    const float* end_t    = (const float*)d_in[9];   // [21]
    const float* trans    = (const float*)d_in[10];  // [21,21]

    float* logits    = (float*)d_out;                // [NTOK, Tn]
    float* out_loss  = logits + (size_t)NTOK * Tn;   // scalar slot
    float* loss_part = (float*)d_ws;                 // [Bn] scratch

    fused_lex_gemm_kernel<<<NTOK / 16, 64, 0, stream>>>(
        seq, lex_ids, lex_wt, word_emb, Wm, bias, logits);

    crf_kernel<<<Bn, 32, 0, stream>>>(
        logits, label, amask, start_t, end_t, trans, loss_part);

    loss_reduce_kernel<<<1, 64, 0, stream>>>(loss_part, out_loss);
}